// PCELayer_32186484916340
// MI455X (gfx1250) — compile-verified
//
#include <hip/hip_runtime.h>
#include <math.h>

#define BB 16
#define CC 128
#define HH 64
#define WW 64
#define HCC 512

typedef float v2f __attribute__((ext_vector_type(2)));
typedef float v8f __attribute__((ext_vector_type(8)));
typedef unsigned v4u __attribute__((ext_vector_type(4)));
typedef unsigned v8u __attribute__((ext_vector_type(8)));

// ---------------------------------------------------------------------------
// Tensor Data Mover: 2D tile DMA (global -> LDS), 4-byte elements.
// Builds D# groups 0/1 per CDNA5 ISA (CDNA5 ISA) and issues
// tensor_load_to_lds via inline asm (portable across both toolchains).
//   tileD0 : contiguous elements per row  (tile_dim0 == tensor_dim0)
//   tileD1 : number of rows              (tile_dim1 == tensor_dim1)
//   stride0: row stride in elements      (tensor_dim0_stride)
// ---------------------------------------------------------------------------
__device__ __forceinline__ void tdm_load_2d(unsigned ldsOff, const void* gptr,
                                            unsigned tileD0, unsigned tileD1,
                                            unsigned stride0) {
  unsigned long long ga = (unsigned long long)gptr;
  v4u g0;
  g0[0] = 1u;                                   // count=1, user descriptor
  g0[1] = ldsOff;                               // lds_addr (bytes)
  g0[2] = (unsigned)ga;                         // global_addr[31:0]
  g0[3] = ((unsigned)(ga >> 32) & 0x01FFFFFFu)  // global_addr[56:32]
          | 0x80000000u;                        // type=2 ("image")
  v8u g1;
  g1[0] = 0x00020000u;                          // data_size=2 (4B), no mask/flags
  g1[1] = (tileD0 & 0xFFFFu) << 16;             // tensor_dim0 lo16
  g1[2] = (tileD0 >> 16) | ((tileD1 & 0xFFFFu) << 16); // dim0 hi16 | tensor_dim1 lo16
  g1[3] = (tileD1 >> 16) | (tileD0 << 16);      // dim1 hi16 | tile_dim0
  g1[4] = tileD1 & 0xFFFFu;                     // tile_dim1 | tile_dim2=0
  g1[5] = stride0;                              // tensor_dim0_stride lo32
  g1[6] = 0u;                                   // stride0 hi16 | dim1_stride lo16
  g1[7] = 0u;
  asm volatile("tensor_load_to_lds %0, %1" :: "s"(g0), "s"(g1) : "memory");
}

// ---------------------------------------------------------------------------
// Router gate: Fourier features of 4x4 patch centers -> 32->64 silu -> 64->8 softmax
// ---------------------------------------------------------------------------
__global__ void gate_kernel(const float* __restrict__ w1, const float* __restrict__ b1,
                            const float* __restrict__ w2, const float* __restrict__ b2,
                            float* __restrict__ wts) {
  int p = threadIdx.x;
  if (p >= 16) return;
  int py = p >> 2, px = p & 3;
  float cy = (py + 0.5f) / 4.0f, cx = (px + 0.5f) / 4.0f;
  float feats[32];
#pragma unroll
  for (int f = 0; f < 8; ++f) {
    float fr = exp2f((float)f) * 3.14159265358979323846f;
    feats[f]      = sinf(cy * fr);
    feats[8 + f]  = cosf(cy * fr);
    feats[16 + f] = sinf(cx * fr);
    feats[24 + f] = cosf(cx * fr);
  }
  float hbuf[64];
  for (int j = 0; j < 64; ++j) {
    float v = b1[j];
    for (int f = 0; f < 32; ++f) v += feats[f] * w1[f * 64 + j];
    hbuf[j] = v / (1.f + expf(-v));
  }
  float lg[8];
  float mx = -1e30f;
  for (int k = 0; k < 8; ++k) {
    float v = b2[k];
    for (int j = 0; j < 64; ++j) v += hbuf[j] * w2[j * 8 + k];
    lg[k] = v;
    mx = fmaxf(mx, v);
  }
  float den = 0.f;
  for (int k = 0; k < 8; ++k) { lg[k] = expf(lg[k] - mx); den += lg[k]; }
  for (int k = 0; k < 8; ++k) wts[p * 8 + k] = lg[k] / den;
}

// ---------------------------------------------------------------------------
// Expert 3x3 conv as implicit GEMM on V_WMMA_F32_16X16X4_F32.
// WG = 256 thr = 8 waves; WG tile = 64 co x 32 pixels.  Weight tile staged via
// TDM (tensor_load_to_lds): 64 rows x 72 contiguous floats, row stride 1152.
// LDS sW layout = TDM fill order: [co64][ci8][tap9].
// Wave (mt,nt) owns a 16x16 tile; 9 taps x 2 K4-steps per ci-block => 288 WMMA.
// GroupNorm sum/sumsq accumulated per (b, group) via wave reduce + atomics.
// ---------------------------------------------------------------------------
__global__ void conv3x3_wmma(const float* __restrict__ x, const float* __restrict__ wexp,
                             float* __restrict__ y, float* __restrict__ statsAcc, int e) {
  __shared__ float sW[4608];  // [co64][ci8][tap9]  (TDM destination)
  __shared__ float sX[816];   // [ci8][row3][col34] (haloed)
  int tid = threadIdx.x;
  int lane = tid & 31, wv = tid >> 5;
  int mt = wv >> 1, nt = wv & 1;
  int hl = lane >> 4, l16 = lane & 15;

  int idx = blockIdx.x;
  int wb = idx & 1; idx >>= 1;
  int h = idx % HH; idx /= HH;
  int b = idx % BB; idx /= BB;
  int cb = idx;  // 0..1 (co block of 64)
  int co0 = cb * 64;
  int w0 = wb * 32;

  v8f c = {};
  const float* wbase = wexp + (size_t)e * CC * CC * 9;

  for (int ci0 = 0; ci0 < CC; ci0 += 8) {
    // --- async DMA of the weight tile via the Tensor Data Mover ---
    if (tid == 0)
      tdm_load_2d((unsigned)(unsigned long long)(void*)sW,
                  wbase + ((size_t)co0 * CC + ci0) * 9,
                  72u, 64u, (unsigned)(CC * 9));
    // --- cooperative haloed activation tile load (overlaps with the DMA) ---
    for (int i = tid; i < 816; i += 256) {
      int cil = i / 102;
      int r = i % 102;
      int row = r / 34, col = r % 34;
      int gh = h + row - 1;
      int gw = w0 + col - 1;
      float v = 0.f;
      if (gh >= 0 && gh < HH && gw >= 0 && gw < WW)
        v = x[(((size_t)b * CC + ci0 + cil) * HH + gh) * WW + gw];
      sX[i] = v;
    }
    // prefetch next ci-block of x into the cache hierarchy
    if (ci0 + 8 < CC)
      __builtin_prefetch(&x[(((size_t)b * CC + (ci0 + 8)) * HH + h) * WW + w0], 0, 1);
    if (tid == 0) __builtin_amdgcn_s_wait_tensorcnt(0);
    __syncthreads();
#pragma unroll
    for (int kh = 0; kh < 3; ++kh) {
#pragma unroll
      for (int kw = 0; kw < 3; ++kw) {
        int tap = kh * 3 + kw;
#pragma unroll
        for (int cio = 0; cio < 8; cio += 4) {
          v2f a, bf;
          int arow = (mt * 16 + l16) * 72 + (cio + hl * 2) * 9 + tap;
          a.x = sW[arow];
          a.y = sW[arow + 9];
          int colx = nt * 16 + l16 + kw;
          bf.x = sX[((cio + hl) * 3 + kh) * 34 + colx];
          bf.y = sX[((cio + 2 + hl) * 3 + kh) * 34 + colx];
          c = __builtin_amdgcn_wmma_f32_16x16x4_f32(false, a, false, bf,
                                                    (short)0, c, false, false);
        }
      }
    }
    __syncthreads();
  }

  float s1 = 0.f, s2 = 0.f;
  int wout = w0 + nt * 16 + l16;
#pragma unroll
  for (int r = 0; r < 8; ++r) {
    float v = c[r];
    s1 += v;
    s2 += v * v;
    int co = co0 + mt * 16 + r + hl * 8;
    y[(((size_t)b * CC + co) * HH + h) * WW + wout] = v;
  }
#pragma unroll
  for (int off = 16; off > 0; off >>= 1) {
    s1 += __shfl_xor(s1, off, 32);
    s2 += __shfl_xor(s2, off, 32);
  }
  if (lane == 0) {
    int g = cb * 4 + mt;  // 16 channels per group == one wave tile
    atomicAdd(&statsAcc[(b * 8 + g) * 2 + 0], s1);
    atomicAdd(&statsAcc[(b * 8 + g) * 2 + 1], s2);
  }
}

// ---------------------------------------------------------------------------
// Generic 1x1-conv GEMM on WMMA f32 (pw1 stride 1, pw2 stride 2 gather).
// Weight tile (64 x 8, row stride K) staged via TDM; LDS layout [m64][k8]
// matches the TDM fill order exactly.
// ---------------------------------------------------------------------------
__global__ void pw_gemm_wmma(const float* __restrict__ Wm, const float* __restrict__ X,
                             float* __restrict__ Y, int M, int K, int Hin, int Win,
                             int OH, int OW, int stride, int owb,
                             float* __restrict__ statsAcc, int gshift) {
  __shared__ float sW[512];  // [m64][k8]  (TDM destination)
  __shared__ float sX[256];  // [k8][n32]
  int tid = threadIdx.x;
  int lane = tid & 31, wv = tid >> 5;
  int mt = wv >> 1, nt = wv & 1;
  int hl = lane >> 4, l16 = lane & 15;

  int idx = blockIdx.x;
  int wi = idx % owb; idx /= owb;
  int oh = idx % OH; idx /= OH;
  int b = idx % BB; idx /= BB;
  int m0 = idx * 64;
  int ow0 = wi * 32;

  v8f c = {};
  for (int k0 = 0; k0 < K; k0 += 8) {
    if (tid == 0)
      tdm_load_2d((unsigned)(unsigned long long)(void*)sW,
                  Wm + (size_t)m0 * K + k0, 8u, 64u, (unsigned)K);
    {
      int kl = tid >> 5, col = tid & 31;
      sX[tid] = X[(((size_t)b * K + k0 + kl) * Hin + oh * stride) * Win + (ow0 + col) * stride];
    }
    if (tid == 0) __builtin_amdgcn_s_wait_tensorcnt(0);
    __syncthreads();
#pragma unroll
    for (int cio = 0; cio < 8; cio += 4) {
      v2f a, bf;
      int arow = (mt * 16 + l16) * 8 + cio + hl * 2;
      a.x = sW[arow];
      a.y = sW[arow + 1];
      int colx = nt * 16 + l16;
      bf.x = sX[(cio + hl) * 32 + colx];
      bf.y = sX[(cio + 2 + hl) * 32 + colx];
      c = __builtin_amdgcn_wmma_f32_16x16x4_f32(false, a, false, bf,
                                                (short)0, c, false, false);
    }
    __syncthreads();
  }

  float s1 = 0.f, s2 = 0.f;
  int ow = ow0 + nt * 16 + l16;
#pragma unroll
  for (int r = 0; r < 8; ++r) {
    float v = c[r];
    s1 += v;
    s2 += v * v;
    int co = m0 + mt * 16 + r + hl * 8;
    Y[(((size_t)b * M + co) * OH + oh) * OW + ow] = v;
  }
#pragma unroll
  for (int off = 16; off > 0; off >>= 1) {
    s1 += __shfl_xor(s1, off, 32);
    s2 += __shfl_xor(s2, off, 32);
  }
  if (lane == 0) {
    int g = (m0 + mt * 16) >> gshift;
    atomicAdd(&statsAcc[(b * 8 + g) * 2 + 0], s1);
    atomicAdd(&statsAcc[(b * 8 + g) * 2 + 1], s2);
  }
}

// ---------------------------------------------------------------------------
// Depthwise 3x3 (cheap, VALU) with GN stat accumulation.
// Block = 256 thr = 4 rows x 64 cols of one (b, channel).
// ---------------------------------------------------------------------------
__global__ void dw_conv3x3(const float* __restrict__ X, const float* __restrict__ wdw,
                           float* __restrict__ Y, float* __restrict__ statsAcc) {
  __shared__ float r1[8], r2[8];
  int tid = threadIdx.x;
  int idx = blockIdx.x;
  int hb = idx & 15; idx >>= 4;
  int cch = idx % HCC; idx /= HCC;
  int b = idx;
  int h = hb * 4 + (tid >> 6);
  int w = tid & 63;
  float k[9];
#pragma unroll
  for (int i = 0; i < 9; ++i) k[i] = wdw[cch * 9 + i];
  const float* xp = X + ((size_t)b * HCC + cch) * (HH * WW);
  float acc = 0.f;
#pragma unroll
  for (int kh = 0; kh < 3; ++kh) {
    int gh = h + kh - 1;
    if (gh < 0 || gh >= HH) continue;
#pragma unroll
    for (int kw = 0; kw < 3; ++kw) {
      int gw = w + kw - 1;
      if (gw < 0 || gw >= WW) continue;
      acc += k[kh * 3 + kw] * xp[gh * WW + gw];
    }
  }
  Y[((size_t)b * HCC + cch) * (HH * WW) + h * WW + w] = acc;
  float s1 = acc, s2 = acc * acc;
#pragma unroll
  for (int off = 16; off > 0; off >>= 1) {
    s1 += __shfl_xor(s1, off, 32);
    s2 += __shfl_xor(s2, off, 32);
  }
  int lane = tid & 31, wv = tid >> 5;
  if (lane == 0) { r1[wv] = s1; r2[wv] = s2; }
  __syncthreads();
  if (tid == 0) {
    float t1 = 0.f, t2 = 0.f;
    for (int i = 0; i < 8; ++i) { t1 += r1[i]; t2 += r2[i]; }
    int g = cch >> 6;
    atomicAdd(&statsAcc[(b * 8 + g) * 2 + 0], t1);
    atomicAdd(&statsAcc[(b * 8 + g) * 2 + 1], t2);
  }
}

// ---------------------------------------------------------------------------
// GN stat helpers + fused epilogues
// ---------------------------------------------------------------------------
__global__ void zero_stats(float* __restrict__ p) { p[threadIdx.x] = 0.f; }

__global__ void finalize_stats(const float* __restrict__ acc, float* __restrict__ mr,
                               float invCount) {
  int i = threadIdx.x;
  if (i < 128) {
    float m = acc[i * 2] * invCount;
    float v = acc[i * 2 + 1] * invCount - m * m;
    mr[i * 2] = m;
    mr[i * 2 + 1] = rsqrtf(v + 1e-5f);
  }
}

// combined = x + sum_e w_e * silu(GN_e(y_e))   (softmax weights sum to 1)
__global__ void accumulate_expert(const float* __restrict__ y, const float* __restrict__ x,
                                  float* __restrict__ cmb, const float* __restrict__ mr,
                                  const float* __restrict__ gs, const float* __restrict__ gb,
                                  const float* __restrict__ wts, int e) {
  size_t i = (size_t)blockIdx.x * 256 + threadIdx.x;
  int pix = (int)(i & 4095);
  int cpl = (int)((i >> 12) & 127);
  int b = (int)(i >> 19);
  int h = pix >> 6, w = pix & 63;
  int g = cpl >> 4;
  float m = mr[(b * 8 + g) * 2], rs = mr[(b * 8 + g) * 2 + 1];
  float v = (y[i] - m) * rs * gs[cpl] + gb[cpl];
  float sv = v / (1.f + __expf(-v));
  float wgt = wts[((h >> 4) * 4 + (w >> 4)) * 8 + e];
  float term = wgt * sv;
  if (e == 0)
    cmb[i] = x[i] + term;
  else
    cmb[i] += term;
}

__global__ void apply_gn_silu(float* __restrict__ d, const float* __restrict__ s,
                              const float* __restrict__ bi, const float* __restrict__ mr,
                              int C, int HW, int gshift) {
  size_t i = (size_t)blockIdx.x * 256 + threadIdx.x;
  int c = (int)((i / HW) % C);
  int b = (int)(i / ((size_t)HW * C));
  int g = c >> gshift;
  float m = mr[(b * 8 + g) * 2], rs = mr[(b * 8 + g) * 2 + 1];
  float v = (d[i] - m) * rs * s[c] + bi[c];
  d[i] = v / (1.f + __expf(-v));
}

// ---------------------------------------------------------------------------
extern "C" void kernel_launch(void* const* d_in, const int* in_sizes, int n_in,
                              void* d_out, int out_size, void* d_ws, size_t ws_size,
                              hipStream_t stream) {
  const float* x      = (const float*)d_in[0];
  const float* w_exp  = (const float*)d_in[1];
  const float* gexp_s = (const float*)d_in[2];
  const float* gexp_b = (const float*)d_in[3];
  const float* w1     = (const float*)d_in[4];
  const float* b1     = (const float*)d_in[5];
  const float* w2     = (const float*)d_in[6];
  const float* b2     = (const float*)d_in[7];
  const float* w_pw1  = (const float*)d_in[8];
  const float* gn1_s  = (const float*)d_in[9];
  const float* gn1_b  = (const float*)d_in[10];
  const float* w_dw   = (const float*)d_in[11];
  const float* gn2_s  = (const float*)d_in[12];
  const float* gn2_b  = (const float*)d_in[13];
  const float* w_pw2  = (const float*)d_in[14];
  const float* gn3_s  = (const float*)d_in[15];
  const float* gn3_b  = (const float*)d_in[16];

  float* ws   = (float*)d_ws;
  float* wts  = ws;                  // 128
  float* sacc = ws + 128;            // 256
  float* smr  = ws + 384;            // 256
  float* ybuf = ws + 1024;           // 8,388,608  (one expert conv output)
  float* cmb  = ybuf + 8388608;      // 8,388,608  (combined)
  float* h1   = cmb + 8388608;       // 33,554,432
  float* h2   = h1 + 33554432;       // 33,554,432
  float* outp = (float*)d_out;

  gate_kernel<<<1, 32, 0, stream>>>(w1, b1, w2, b2, wts);

  for (int e = 0; e < 8; ++e) {
    zero_stats<<<1, 256, 0, stream>>>(sacc);
    conv3x3_wmma<<<4096, 256, 0, stream>>>(x, w_exp, ybuf, sacc, e);
    finalize_stats<<<1, 128, 0, stream>>>(sacc, smr, 1.f / 65536.f);
    accumulate_expert<<<32768, 256, 0, stream>>>(ybuf, x, cmb, smr,
                                                 gexp_s + e * 128, gexp_b + e * 128,
                                                 wts, e);
  }

  // pw1: 128 -> 512
  zero_stats<<<1, 256, 0, stream>>>(sacc);
  pw_gemm_wmma<<<16384, 256, 0, stream>>>(w_pw1, cmb, h1, 512, 128, 64, 64, 64, 64, 1, 2,
                                          sacc, 6);
  finalize_stats<<<1, 128, 0, stream>>>(sacc, smr, 1.f / 262144.f);
  apply_gn_silu<<<131072, 256, 0, stream>>>(h1, gn1_s, gn1_b, smr, 512, 4096, 6);

  // depthwise 3x3 on 512 channels
  zero_stats<<<1, 256, 0, stream>>>(sacc);
  dw_conv3x3<<<16 * 512 * 16, 256, 0, stream>>>(h1, w_dw, h2, sacc);
  finalize_stats<<<1, 128, 0, stream>>>(sacc, smr, 1.f / 262144.f);
  apply_gn_silu<<<131072, 256, 0, stream>>>(h2, gn2_s, gn2_b, smr, 512, 4096, 6);

  // pw2 stride 2: 512 -> 128, output 32x32 -> d_out (fully overwritten each call)
  zero_stats<<<1, 256, 0, stream>>>(sacc);
  pw_gemm_wmma<<<1024, 256, 0, stream>>>(w_pw2, h2, outp, 128, 512, 64, 64, 32, 32, 2, 1,
                                         sacc, 4);
  finalize_stats<<<1, 128, 0, stream>>>(sacc, smr, 1.f / 16384.f);
  apply_gn_silu<<<8192, 256, 0, stream>>>(outp, gn3_s, gn3_b, smr, 128, 1024, 4);
}